// Grouping_55001351193100
// MI455X (gfx1250) — compile-verified
//
#include <hip/hip_runtime.h>

typedef float v2f __attribute__((ext_vector_type(2)));
typedef float v8f __attribute__((ext_vector_type(8)));

// Problem constants (from reference): B=16, S=8192, H=512, G=2048
#define BB 16
#define SS 8192
#define HH 512
#define GG 2048

#define GT 16              // groups per tile  (WMMA M)
#define HT 16              // H columns per tile (WMMA N)
#define NGT (GG / GT)      // 128 group tiles per batch row
#define NHT (HH / HT)      // 32 column tiles
#define WAVES_PER_BLOCK 8  // 256 threads, wave32

// One wave computes out[b, g0:g0+16, h0:h0+16] as a dense 16xK * Kx16 matmul
// with A[m,k] = (seg[s]==g0+m) ? 1/count(g0+m) : 0 over the contiguous token
// range owned by the group tile (seg_ids sorted => range is contiguous).
__global__ __launch_bounds__(256) void seg_mean_wmma(
    const float* __restrict__ feats,   // [B,S,H]
    const int*   __restrict__ seg_ids, // [B,S] sorted per row
    float*       __restrict__ out)     // [B,G,H]
{
    const int wave  = blockIdx.x * WAVES_PER_BLOCK + (threadIdx.x >> 5);
    const int lane  = threadIdx.x & 31;
    const int htile = wave & (NHT - 1);          // fastest: waves in a block share tokens
    const int gtile = (wave >> 5) & (NGT - 1);
    const int b     = wave >> 12;                // 32*128 = 4096 jobs per batch row

    const int g0 = gtile * GT;
    const int h0 = htile * HT;
    const int m  = lane & 15;                    // A-row / B-col / D-col for this lane
    const int kb = (lane >> 4) << 1;             // K base: lanes 0-15 -> {0,1}, 16-31 -> {2,3}
    const int gm = g0 + m;

    const int* __restrict__ seg = seg_ids + b * SS;

    // lower_bound(g0+m) and lower_bound(g0+m+1) -> per-group [start,end)
    int t  = gm;
    int lo = 0, hi = SS;
    while (lo < hi) { int mid = (lo + hi) >> 1; if (seg[mid] < t) lo = mid + 1; else hi = mid; }
    const int lb_lo = lo;
    lo = 0; hi = SS; ++t;
    while (lo < hi) { int mid = (lo + hi) >> 1; if (seg[mid] < t) lo = mid + 1; else hi = mid; }
    const int lb_hi = lo;

    const int   cnt = lb_hi - lb_lo;
    const float inv = (cnt > 0) ? (1.0f / (float)cnt) : 0.0f;

    // Wave-uniform token range for the whole 16-group tile (scalar loop bounds
    // => EXEC stays all-ones around the WMMA, as the ISA requires).
    const int sA = __builtin_amdgcn_readfirstlane(lb_lo);                 // lower_bound(g0)
    const int sB = __builtin_amdgcn_readfirstlane(__shfl(lb_hi, 15, 32)); // lower_bound(g0+16)

    v8f c = {0.f, 0.f, 0.f, 0.f, 0.f, 0.f, 0.f, 0.f};
    const float* __restrict__ fb = feats + ((size_t)b * SS) * HH + h0 + m;

    // 8 tokens (2 WMMA K-steps) per iteration. All loads are UNCONDITIONAL
    // (address clamped to a valid token); the range predicate only gates the
    // A-matrix selector via v_cndmask, so no EXEC manipulation and 8 loads
    // are in flight before the first wait.
    for (int s0 = sA; s0 < sB; s0 += 8) {
        int   sgv[4];
        float fv[4];
        int   okv[4];
#pragma unroll
        for (int u = 0; u < 4; ++u) {
            const int s  = s0 + ((u >> 1) << 2) + kb + (u & 1); // step*(4) + kb + j
            const int sc = (s < SS - 1) ? s : (SS - 1);          // always-valid address
            sgv[u] = seg[sc];
            fv[u]  = fb[(size_t)sc * HH];
            okv[u] = (s < sB);
        }
        v2f a0, b0, a1, b1;
#pragma unroll
        for (int j = 0; j < 2; ++j) {
            a0[j] = (okv[j]     && sgv[j]     == gm) ? inv : 0.0f;
            a1[j] = (okv[2 + j] && sgv[2 + j] == gm) ? inv : 0.0f;
            b0[j] = fv[j];
            b1[j] = fv[2 + j];
        }
        // D = A(16x4,f32) x B(4x16,f32) + C ; full f32 precision
        c = __builtin_amdgcn_wmma_f32_16x16x4_f32(false, a0, false, b0,
                                                  (short)0, c, false, false);
        c = __builtin_amdgcn_wmma_f32_16x16x4_f32(false, a1, false, b1,
                                                  (short)0, c, false, false);
    }

    // D layout: lane L -> col n=L&15, VGPR r -> row M = 8*(L>=16) + r.
    // Every out element written exactly once; empty groups naturally write 0.
    // NT stores: output is written once and never re-read -> don't pollute L2.
    const int mb = (lane >> 4) << 3;
    float* __restrict__ ob = out + (((size_t)b * GG) + g0 + mb) * HH + h0 + m;
#pragma unroll
    for (int r = 0; r < 8; ++r)
        __builtin_nontemporal_store(c[r], ob + (size_t)r * HH);
}

extern "C" void kernel_launch(void* const* d_in, const int* in_sizes, int n_in,
                              void* d_out, int out_size, void* d_ws, size_t ws_size,
                              hipStream_t stream) {
    (void)in_sizes; (void)n_in; (void)d_ws; (void)ws_size; (void)out_size;
    const float* feats   = (const float*)d_in[0];
    const int*   seg_ids = (const int*)d_in[1];
    float*       out     = (float*)d_out;

    const int total_waves = BB * NGT * NHT;                // 65536
    const int blocks      = total_waves / WAVES_PER_BLOCK; // 8192
    seg_mean_wmma<<<blocks, 256, 0, stream>>>(feats, seg_ids, out);
}